// TwoLayerAttGCNWithPooling_2233382994099
// MI455X (gfx1250) — compile-verified
//
#include <hip/hip_runtime.h>
#include <math.h>

#define BG   128
#define NPG  512
#define NN   (BG * NPG)      // 65536 nodes
#define DEG  16
#define EE   (NN * DEG)      // 1048576 edges
#define FF   64

typedef __attribute__((ext_vector_type(2))) float v2f;
typedef __attribute__((ext_vector_type(8))) float v8f;

// ---------------------------------------------------------------- utilities
__device__ inline void atomicMaxFloat(float* addr, float val) {
  // sign-split trick: int ordering == float ordering for +, reversed unsigned for -
  if (val >= 0.f) atomicMax((int*)addr, __float_as_int(val));
  else            atomicMin((unsigned int*)addr, __float_as_uint(val));
}

// ---------------------------------------------------------------- edge init
__global__ void init_edges_k(const int* __restrict__ ei, int* __restrict__ srcb,
                             int* __restrict__ dstb, int* __restrict__ maskb) {
  int i = blockIdx.x * blockDim.x + threadIdx.x;
  if (i >= EE) return;
  srcb[i]  = ei[i];
  dstb[i]  = ei[EE + i];
  maskb[i] = 1;
}

// ---------------------------------------------------------------- fp32 WMMA GEMM: H[n,64] = X[n,64] @ W[64,64]
// 1 wave -> 16 rows x 64 cols. 4 waves/block -> 64 rows/block. n is a multiple of 64.
__global__ __launch_bounds__(128)
void gemm_n64_wmma(const float* __restrict__ X, const float* __restrict__ W,
                   float* __restrict__ H, int n) {
  const int wave = threadIdx.x >> 5;
  const int lane = threadIdx.x & 31;
  const int row0 = blockIdx.x * 64 + wave * 16;
  if (row0 >= n) return;                    // uniform per wave: EXEC stays all-ones
  const int half = lane >> 4;               // 0: K lo pair, 1: K hi pair
  const int l15  = lane & 15;

  v8f acc0 = {}, acc1 = {}, acc2 = {}, acc3 = {};
  const float* xrow = X + (size_t)(row0 + l15) * FF;

  #pragma unroll
  for (int kk = 0; kk < 16; ++kk) {
    const int kbase = kk * 4;
    // A 16x4 fp32 fragment: lane(l15,half) holds A[l15][kbase + 2*half + {0,1}]
    v2f a;
    a.x = xrow[kbase + 2 * half + 0];
    a.y = xrow[kbase + 2 * half + 1];
    // B 4x16 fp32 fragments: lane holds B[kbase + 2*half + v][col0 + l15]
    const float* wr0 = W + (size_t)(kbase + 2 * half) * FF + l15;
    const float* wr1 = wr0 + FF;
    v2f b0, b1, b2, b3;
    b0.x = wr0[0];   b0.y = wr1[0];
    b1.x = wr0[16];  b1.y = wr1[16];
    b2.x = wr0[32];  b2.y = wr1[32];
    b3.x = wr0[48];  b3.y = wr1[48];
    acc0 = __builtin_amdgcn_wmma_f32_16x16x4_f32(false, a, false, b0, (short)0, acc0, false, false);
    acc1 = __builtin_amdgcn_wmma_f32_16x16x4_f32(false, a, false, b1, (short)0, acc1, false, false);
    acc2 = __builtin_amdgcn_wmma_f32_16x16x4_f32(false, a, false, b2, (short)0, acc2, false, false);
    acc3 = __builtin_amdgcn_wmma_f32_16x16x4_f32(false, a, false, b3, (short)0, acc3, false, false);
  }
  // C/D layout: VGPR r -> row row0 + r + 8*half, col = l15 (+tile offset)
  #pragma unroll
  for (int r = 0; r < 8; ++r) {
    float* hr = H + (size_t)(row0 + r + 8 * half) * FF + l15;
    hr[0]  = acc0[r];
    hr[16] = acc1[r];
    hr[32] = acc2[r];
    hr[48] = acc3[r];
  }
}

// ---------------------------------------------------------------- al_s/al_d = h . a  (warp per node, wave32)
__global__ void rowdot2_k(const float* __restrict__ H, const float* __restrict__ as,
                          const float* __restrict__ ad, float* __restrict__ alS,
                          float* __restrict__ alD, int n) {
  const int lane = threadIdx.x & 31;
  const int node = blockIdx.x * (blockDim.x >> 5) + (threadIdx.x >> 5);
  if (node >= n) return;
  const float h0 = H[(size_t)node * FF + lane];
  const float h1 = H[(size_t)node * FF + 32 + lane];
  float vs = h0 * as[lane] + h1 * as[32 + lane];
  float vd = h0 * ad[lane] + h1 * ad[32 + lane];
  #pragma unroll
  for (int off = 16; off > 0; off >>= 1) {
    vs += __shfl_xor(vs, off);
    vd += __shfl_xor(vd, off);
  }
  if (lane == 0) { alS[node] = vs; alD[node] = vd; }
}

// ---------------------------------------------------------------- per-layer init: zero G, init emax/denom
__global__ void init_layer_k(float* __restrict__ G, float* __restrict__ emax,
                             float* __restrict__ denom, int n) {
  int i = blockIdx.x * blockDim.x + threadIdx.x;
  if (i < n * FF) G[i] = 0.f;
  if (i < n) { emax[i] = __int_as_float((int)0xFF7FFFFF); denom[i] = 0.f; }   // -FLT_MAX
}

// ---------------------------------------------------------------- attention pass 1: e = lrelu, segment max
// items: [0,EE) real edges, [EE, EE+n) self loops
__global__ void attn_pass1_k(const int* __restrict__ srcb, const int* __restrict__ dstb,
                             const int* __restrict__ maskb, const float* __restrict__ alS,
                             const float* __restrict__ alD, float* __restrict__ emax,
                             float* __restrict__ pbuf, int n) {
  const int id = blockIdx.x * blockDim.x + threadIdx.x;
  const int total = EE + n;
  if (id >= total) return;
  int s, d, m;
  if (id < EE) { s = srcb[id]; d = dstb[id]; m = maskb[id]; }
  else         { s = d = id - EE; m = 1; }
  const float v = alS[s] + alD[d];
  const float e = v > 0.f ? v : 0.2f * v;   // leaky_relu(0.2)
  pbuf[id] = e;
  if (m) atomicMaxFloat(&emax[d], e);
}

// ---------------------------------------------------------------- attention pass 2: p = exp(e - max), denom
__global__ void attn_pass2_k(const int* __restrict__ dstb, const int* __restrict__ maskb,
                             const float* __restrict__ emax, float* __restrict__ pbuf,
                             float* __restrict__ denom, int n) {
  const int id = blockIdx.x * blockDim.x + threadIdx.x;
  const int total = EE + n;
  if (id >= total) return;
  int d, m;
  if (id < EE) { d = dstb[id]; m = maskb[id]; }
  else         { d = id - EE; m = 1; }
  float p = 0.f;
  if (m) p = expf(pbuf[id] - emax[d]);
  pbuf[id] = p;
  if (p != 0.f) atomicAdd(&denom[d], p);
}

// ---------------------------------------------------------------- attention pass 3: G[dst] += alpha * H[src]  (warp per item)
__global__ void attn_pass3_k(const int* __restrict__ srcb, const int* __restrict__ dstb,
                             const float* __restrict__ pbuf, const float* __restrict__ denom,
                             const float* __restrict__ H, float* __restrict__ G, int n) {
  const int lane = threadIdx.x & 31;
  const int item = blockIdx.x * (blockDim.x >> 5) + (threadIdx.x >> 5);
  const int total = EE + n;
  if (item >= total) return;
  int s, d;
  if (item < EE) { s = srcb[item]; d = dstb[item]; }
  else           { s = d = item - EE; }
  const float p = pbuf[item];
  if (p == 0.f) return;                      // masked or zero-weight edge
  const float alpha = p / denom[d];
  const float v0 = H[(size_t)s * FF + lane] * alpha;
  const float v1 = H[(size_t)s * FF + 32 + lane] * alpha;
  atomicAdd(&G[(size_t)d * FF + lane], v0);
  atomicAdd(&G[(size_t)d * FF + 32 + lane], v1);
}

// ---------------------------------------------------------------- G += bias (broadcast)
__global__ void bias_add_k(float* __restrict__ G, const float* __restrict__ b, int n) {
  int i = blockIdx.x * blockDim.x + threadIdx.x;
  if (i < n * FF) G[i] += b[i & (FF - 1)];
}

// ---------------------------------------------------------------- ||w|| (trivial)
__global__ void norm64_k(const float* __restrict__ w, float* __restrict__ out) {
  if (threadIdx.x == 0 && blockIdx.x == 0) {
    float s = 0.f;
    for (int i = 0; i < FF; ++i) s += w[i] * w[i];
    out[0] = sqrtf(s);
  }
}

// ---------------------------------------------------------------- score = sigmoid((G.w)/||w||)  (warp per node)
__global__ void score_k(const float* __restrict__ G, const float* __restrict__ w,
                        const float* __restrict__ wn, float* __restrict__ score, int n) {
  const int lane = threadIdx.x & 31;
  const int node = blockIdx.x * (blockDim.x >> 5) + (threadIdx.x >> 5);
  if (node >= n) return;
  float v = G[(size_t)node * FF + lane] * w[lane] + G[(size_t)node * FF + 32 + lane] * w[32 + lane];
  #pragma unroll
  for (int off = 16; off > 0; off >>= 1) v += __shfl_xor(v, off);
  if (lane == 0) {
    const float x = v / wn[0];
    score[node] = 1.f / (1.f + expf(-x));
  }
}

// ---------------------------------------------------------------- per-graph top-k by rank counting (block per graph)
__global__ void topk_k(const float* __restrict__ score, int* __restrict__ newid,
                       int n_per, int k) {
  __shared__ float s[512];
  const int g = blockIdx.x;
  const int i = threadIdx.x;            // blockDim.x == n_per
  const float mine = score[g * n_per + i];
  s[i] = mine;
  __syncthreads();
  int rank = 0;
  for (int j = 0; j < n_per; ++j) {
    const float o = s[j];
    rank += (o > mine) || (o == mine && j < i);   // jax.top_k tie-break: lower index first
  }
  newid[g * n_per + i] = (rank < k) ? (g * k + rank) : -1;
}

// ---------------------------------------------------------------- gather kept rows, gate by score, relu (warp per old node)
__global__ void pool_gather_k(const float* __restrict__ G, const float* __restrict__ score,
                              const int* __restrict__ newid, float* __restrict__ Xn, int n_old) {
  const int lane = threadIdx.x & 31;
  const int node = blockIdx.x * (blockDim.x >> 5) + (threadIdx.x >> 5);
  if (node >= n_old) return;
  const int nid = newid[node];
  if (nid < 0) return;
  const float sc = score[node];
  const float v0 = G[(size_t)node * FF + lane] * sc;
  const float v1 = G[(size_t)node * FF + 32 + lane] * sc;
  Xn[(size_t)nid * FF + lane]      = fmaxf(v0, 0.f);
  Xn[(size_t)nid * FF + 32 + lane] = fmaxf(v1, 0.f);
}

// ---------------------------------------------------------------- relabel edges in place
__global__ void relabel_k(int* __restrict__ srcb, int* __restrict__ dstb,
                          int* __restrict__ maskb, const int* __restrict__ newid) {
  const int i = blockIdx.x * blockDim.x + threadIdx.x;
  if (i >= EE) return;
  const int ns = newid[srcb[i]];
  const int nd = newid[dstb[i]];
  maskb[i] = maskb[i] & (ns >= 0) & (nd >= 0);
  srcb[i]  = ns < 0 ? 0 : ns;
  dstb[i]  = nd < 0 ? 0 : nd;
}

// ---------------------------------------------------------------- readout: gmp/gap -> xcat[B, 2F]  (block per graph)
__global__ void readout_k(const float* __restrict__ H, float* __restrict__ xcat, int n_per) {
  const int g = blockIdx.x;
  const int f = threadIdx.x;            // 64 threads
  float mx = -3.402823466e38f, sm = 0.f;
  for (int node = 0; node < n_per; ++node) {
    const float v = H[((size_t)g * n_per + node) * FF + f];
    mx = fmaxf(mx, v);
    sm += v;
  }
  xcat[g * (2 * FF) + f]      = mx;
  xcat[g * (2 * FF) + FF + f] = sm / (float)n_per;
}

// ---------------------------------------------------------------- head: out = sigmoid(xcat @ Wo + bo)
__global__ void head_k(const float* __restrict__ xcat, const float* __restrict__ Wo,
                       const float* __restrict__ bo, float* __restrict__ out) {
  const int g = blockIdx.x * blockDim.x + threadIdx.x;
  if (g >= BG) return;
  float acc = bo[0];
  for (int j = 0; j < 2 * FF; ++j) acc += xcat[g * (2 * FF) + j] * Wo[j];
  out[g] = 1.f / (1.f + expf(-acc));
}

// ================================================================ launch
extern "C" void kernel_launch(void* const* d_in, const int* in_sizes, int n_in,
                              void* d_out, int out_size, void* d_ws, size_t ws_size,
                              hipStream_t stream) {
  (void)in_sizes; (void)n_in; (void)out_size; (void)ws_size;

  const float* x  = (const float*)d_in[0];
  const int*   ei = (const int*)d_in[1];
  // d_in[2] edge_weight: unused by model; d_in[3] batch_index: implied by layout
  const float* Wl[3] = {(const float*)d_in[4],  (const float*)d_in[9],  (const float*)d_in[14]};
  const float* As[3] = {(const float*)d_in[5],  (const float*)d_in[10], (const float*)d_in[15]};
  const float* Ad[3] = {(const float*)d_in[6],  (const float*)d_in[11], (const float*)d_in[16]};
  const float* Bb[3] = {(const float*)d_in[7],  (const float*)d_in[12], (const float*)d_in[17]};
  const float* Pw[3] = {(const float*)d_in[8],  (const float*)d_in[13], (const float*)d_in[18]};
  const float* Wo = (const float*)d_in[19];
  const float* bo = (const float*)d_in[20];

  float* out  = (float*)d_out;          // [B]
  float* xcat = out + BG;               // [B, 2F]

  // workspace carve (~58 MB total)
  char* base = (char*)d_ws;
  float* h     = (float*)base; base += (size_t)NN * FF * 4;        // linear features
  float* g     = (float*)base; base += (size_t)NN * FF * 4;        // GAT output
  float* xb    = (float*)base; base += (size_t)(NN / 2) * FF * 4;  // pooled features
  float* alS   = (float*)base; base += (size_t)NN * 4;
  float* alD   = (float*)base; base += (size_t)NN * 4;
  float* emax  = (float*)base; base += (size_t)NN * 4;
  float* denom = (float*)base; base += (size_t)NN * 4;
  float* score = (float*)base; base += (size_t)NN * 4;
  float* pbuf  = (float*)base; base += (size_t)(EE + NN) * 4;
  int*   newid = (int*)base;   base += (size_t)NN * 4;
  int*   srcb  = (int*)base;   base += (size_t)EE * 4;
  int*   dstb  = (int*)base;   base += (size_t)EE * 4;
  int*   maskb = (int*)base;   base += (size_t)EE * 4;
  float* wn    = (float*)base;

  init_edges_k<<<EE / 256, 256, 0, stream>>>(ei, srcb, dstb, maskb);

  const float* curx = x;
  int n = NN, n_per = NPG;
  for (int l = 0; l < 3; ++l) {
    // linear (fp32 WMMA)
    gemm_n64_wmma<<<n / 64, 128, 0, stream>>>(curx, Wl[l], h, n);
    // attention logits per node
    rowdot2_k<<<(n + 7) / 8, 256, 0, stream>>>(h, As[l], Ad[l], alS, alD, n);
    init_layer_k<<<(n * FF + 255) / 256, 256, 0, stream>>>(g, emax, denom, n);
    // segment softmax + scatter-add (items = E edges + n self loops)
    const int items = EE + n;
    attn_pass1_k<<<(items + 255) / 256, 256, 0, stream>>>(srcb, dstb, maskb, alS, alD, emax, pbuf, n);
    attn_pass2_k<<<(items + 255) / 256, 256, 0, stream>>>(dstb, maskb, emax, pbuf, denom, n);
    attn_pass3_k<<<(items + 7) / 8, 256, 0, stream>>>(srcb, dstb, pbuf, denom, h, g, n);
    bias_add_k<<<(n * FF + 255) / 256, 256, 0, stream>>>(g, Bb[l], n);
    // top-k pooling
    norm64_k<<<1, 64, 0, stream>>>(Pw[l], wn);
    score_k<<<(n + 7) / 8, 256, 0, stream>>>(g, Pw[l], wn, score, n);
    const int k = n_per / 2;
    topk_k<<<BG, n_per, 0, stream>>>(score, newid, n_per, k);
    pool_gather_k<<<(n + 7) / 8, 256, 0, stream>>>(g, score, newid, xb, n);
    relabel_k<<<(EE + 255) / 256, 256, 0, stream>>>(srcb, dstb, maskb, newid);
    n = BG * k;
    n_per = k;
    curx = xb;
  }

  // readout + head: out = sigmoid([gmp|gap] @ Wo + bo)
  readout_k<<<BG, 64, 0, stream>>>(xb, xcat, n_per);
  head_k<<<1, BG, 0, stream>>>(xcat, Wo, bo, out);
}